// Block1_87144886436577
// MI455X (gfx1250) — compile-verified
//
#include <hip/hip_runtime.h>
#include <hip/hip_bf16.h>
#include <math.h>

// Problem constants (from reference)
#define NB    2     // batch
#define CIN   3
#define HIN   32
#define CMID  32
#define HMID  16
#define COUT  64
#define HOUT  8
#define NPOS  64    // HOUT*HOUT
#define NEMB  512
#define DIM   64    // == COUT

typedef float v2f __attribute__((ext_vector_type(2)));
typedef float v8f __attribute__((ext_vector_type(8)));

// ---------------- conv1: y1 = relu(conv(x,w1,stride2,pad1)+b1) -> [B][32][16][16]
__global__ void k_conv1(const float* __restrict__ x, const float* __restrict__ w1,
                        const float* __restrict__ b1, float* __restrict__ y1) {
  int t = blockIdx.x * blockDim.x + threadIdx.x;          // NB*32*16*16 = 16384
  int pw = t & 15, ph = (t >> 4) & 15, m = (t >> 8) & 31, b = t >> 13;
  float acc = b1[m];
  for (int ci = 0; ci < CIN; ++ci)
    for (int kh = 0; kh < 4; ++kh) {
      int ih = ph * 2 - 1 + kh;
      if ((unsigned)ih >= 32u) continue;
      for (int kw = 0; kw < 4; ++kw) {
        int iw = pw * 2 - 1 + kw;
        if ((unsigned)iw >= 32u) continue;
        acc += x[((b * CIN + ci) * 32 + ih) * 32 + iw] *
               w1[((m * CIN + ci) * 4 + kh) * 4 + kw];
      }
    }
  y1[t] = fmaxf(acc, 0.f);
}

// ---------------- conv2: yf[b][pos][c] = relu(conv(y1,w2)+b2)
__global__ void k_conv2(const float* __restrict__ y1, const float* __restrict__ w2,
                        const float* __restrict__ b2, float* __restrict__ yf) {
  int t = blockIdx.x * blockDim.x + threadIdx.x;          // NB*64*64 = 8192
  int c = t & 63, pos = (t >> 6) & 63, b = t >> 12;
  int oh = pos >> 3, ow = pos & 7;
  float acc = b2[c];
  for (int m = 0; m < CMID; ++m)
    for (int kh = 0; kh < 4; ++kh) {
      int ph = oh * 2 - 1 + kh;
      if ((unsigned)ph >= 16u) continue;
      for (int kw = 0; kw < 4; ++kw) {
        int pw = ow * 2 - 1 + kw;
        if ((unsigned)pw >= 16u) continue;
        acc += y1[((b * CMID + m) * 16 + ph) * 16 + pw] *
               w2[((c * CMID + m) * 4 + kh) * 4 + kw];
      }
    }
  yf[t] = fmaxf(acc, 0.f);
}

// ---------------- scores: S[b][pos][n] = beta * dot(Y[b][pos], K[n])  (WMMA f32)
// one wave per 16x16 tile; M=64 pos, N=512 keys, K=64
__global__ void k_scores(const float* __restrict__ Ym, const float* __restrict__ Km,
                         float* __restrict__ S, float beta) {
  int wid  = (blockIdx.x * blockDim.x + threadIdx.x) >> 5;   // 0..255
  int lane = threadIdx.x & 31;
  int nt = wid & 31, mt = (wid >> 5) & 3, b = wid >> 7;
  int m0 = mt * 16, n0 = nt * 16;
  const float* Yb = Ym + b * (NPOS * DIM);
  int row_a = m0 + (lane & 15);
  int row_b = n0 + (lane & 15);
  int koff  = 2 * (lane >> 4);
  v8f acc = {};
  for (int k0 = 0; k0 < DIM; k0 += 4) {
    int kk = k0 + koff;
    v2f a, bb;
    a[0]  = Yb[row_a * DIM + kk];  a[1]  = Yb[row_a * DIM + kk + 1];
    bb[0] = Km[row_b * DIM + kk];  bb[1] = Km[row_b * DIM + kk + 1];
    acc = __builtin_amdgcn_wmma_f32_16x16x4_f32(false, a, false, bb,
                                                (short)0, acc, false, false);
  }
  float* Sb = S + b * (NPOS * NEMB);
  int col   = n0 + (lane & 15);
  int rbase = m0 + 8 * (lane >> 4);
#pragma unroll
  for (int j = 0; j < 8; ++j)
    Sb[(rbase + j) * NEMB + col] = acc[j] * beta;
}

// ---------------- softmax over rows of length 512, one wave per row, in place
__global__ void k_softmax(float* __restrict__ S, int nrows) {
  int wid  = (blockIdx.x * blockDim.x + threadIdx.x) >> 5;
  int lane = threadIdx.x & 31;
  if (wid >= nrows) return;
  float* row = S + (size_t)wid * NEMB;
  float mx = -3.402823466e38f;
  for (int j = lane; j < NEMB; j += 32) mx = fmaxf(mx, row[j]);
  for (int off = 16; off > 0; off >>= 1) mx = fmaxf(mx, __shfl_xor(mx, off, 32));
  float sum = 0.f;
  for (int j = lane; j < NEMB; j += 32) { float e = expf(row[j] - mx); row[j] = e; sum += e; }
  for (int off = 16; off > 0; off >>= 1) sum += __shfl_xor(sum, off, 32);
  float inv = 1.f / sum;
  for (int j = lane; j < NEMB; j += 32) row[j] *= inv;
}

// ---------------- attn@V: E[b][pos][c]  (WMMA f32). nchw!=0 -> store [b][c][pos]
__global__ void k_attnv(const float* __restrict__ A, const float* __restrict__ V,
                        float* __restrict__ out, int nchw) {
  int wid  = (blockIdx.x * blockDim.x + threadIdx.x) >> 5;   // 0..31
  int lane = threadIdx.x & 31;
  int nt = wid & 3, mt = (wid >> 2) & 3, b = wid >> 4;
  int m0 = mt * 16, n0 = nt * 16;
  const float* Ab = A + b * (NPOS * NEMB);
  int rowm = m0 + (lane & 15);
  int coln = n0 + (lane & 15);
  int koff = 2 * (lane >> 4);
  v8f acc = {};
  for (int k0 = 0; k0 < NEMB; k0 += 4) {
    int kk = k0 + koff;
    v2f a, bb;
    a[0]  = Ab[rowm * NEMB + kk];    a[1]  = Ab[rowm * NEMB + kk + 1];
    bb[0] = V[kk * DIM + coln];      bb[1] = V[(kk + 1) * DIM + coln];
    acc = __builtin_amdgcn_wmma_f32_16x16x4_f32(false, a, false, bb,
                                                (short)0, acc, false, false);
  }
  int rbase = m0 + 8 * (lane >> 4);
#pragma unroll
  for (int j = 0; j < 8; ++j) {
    int pos = rbase + j;
    if (nchw) out[(b * DIM + coln) * NPOS + pos] = acc[j];
    else      out[(b * NPOS + pos) * DIM + coln] = acc[j];
  }
}

// ---------------- g2[b][o][c] = 2*(y - yq) * relu'(z2)  (relu' = y>0)
__global__ void k_grad(const float* __restrict__ yf, const float* __restrict__ yq,
                       float* __restrict__ g2) {
  int t = blockIdx.x * blockDim.x + threadIdx.x;   // 8192
  float y = yf[t];
  g2[t] = (y > 0.f) ? 2.f * (y - yq[t]) : 0.f;
}

// ---------------- e_total rows: one block per (b,o); deterministic, no atomics
__global__ void k_etotal(const float* __restrict__ g2, const float* __restrict__ w2,
                         const float* __restrict__ y1, const float* __restrict__ w1,
                         float* __restrict__ et) {
  __shared__ float g[COUT];
  __shared__ float gm[CMID][16];   // backprop'ed grad at mid layer, 4x4 window of o
  __shared__ float ws1[CMID][16];  // w1 summed over ci per (kh,kw)
  __shared__ float patch[100];     // 10x10 input receptive field
  int blk = blockIdx.x, b = blk >> 6, o = blk & 63;
  int oh = o >> 3, ow = o & 7;
  int tid = threadIdx.x;           // 128 threads
  if (tid < COUT) g[tid] = g2[(b * NPOS + o) * COUT + tid];
  __syncthreads();
  for (int task = tid; task < CMID * 16; task += 128) {
    int m = task >> 4, slot = task & 15;
    int k2h = slot >> 2, k2w = slot & 3;
    int ph = oh * 2 - 1 + k2h, pw = ow * 2 - 1 + k2w;
    float v = 0.f;
    if ((unsigned)ph < 16u && (unsigned)pw < 16u &&
        y1[((b * CMID + m) * 16 + ph) * 16 + pw] > 0.f) {
      for (int c = 0; c < COUT; ++c)
        v += g[c] * w2[((c * CMID + m) * 4 + k2h) * 4 + k2w];
    }
    gm[m][slot] = v;
    int k1h = k2h, k1w = k2w;      // reuse slot as (kh,kw) index for w1-sum
    ws1[m][slot] = w1[((m * CIN + 0) * 4 + k1h) * 4 + k1w] +
                   w1[((m * CIN + 1) * 4 + k1h) * 4 + k1w] +
                   w1[((m * CIN + 2) * 4 + k1h) * 4 + k1w];
  }
  __syncthreads();
  // patch[dh][dw]: dh = 2*k2h + k1h, dw = 2*k2w + k1w; enumerate decompositions
  for (int cell = tid; cell < 100; cell += 128) {
    int dh = cell / 10, dw = cell % 10;
    float v = 0.f;
    for (int k2h = 0; k2h < 4; ++k2h) {
      int k1h = dh - 2 * k2h;
      if ((unsigned)k1h >= 4u) continue;
      for (int k2w = 0; k2w < 4; ++k2w) {
        int k1w = dw - 2 * k2w;
        if ((unsigned)k1w >= 4u) continue;
        int slot = k2h * 4 + k2w, k1 = k1h * 4 + k1w;
        for (int m = 0; m < CMID; ++m) v += gm[m][slot] * ws1[m][k1];
      }
    }
    patch[cell] = v;
  }
  __syncthreads();
  // write full row of 1024 (structural zeros outside receptive field)
  int ih0 = 4 * oh - 3, iw0 = 4 * ow - 3;
  float* row = et + (size_t)(b * NPOS + o) * 1024;
  for (int i = tid; i < 1024; i += 128) {
    int ih = i >> 5, iw = i & 31;
    int dh = ih - ih0, dw = iw - iw0;
    float v = 0.f;
    if ((unsigned)dh < 10u && (unsigned)dw < 10u) v = patch[dh * 10 + dw];
    row[i] = v;
  }
}

// ---------------- sel[b][i] = first argmin over o of e_total[b][o][i]
__global__ void k_argmin(const float* __restrict__ et, int* __restrict__ sel) {
  int t = blockIdx.x * blockDim.x + threadIdx.x;   // NB*1024
  int b = t >> 10, i = t & 1023;
  const float* base = et + (size_t)b * NPOS * 1024 + i;
  float best = base[0];
  int bi = 0;
  for (int o = 1; o < NPOS; ++o) {
    float v = base[o * 1024];
    if (v < best) { best = v; bi = o; }
  }
  sel[t] = bi;
}

// ---------------- y_masked[b][o][c]: linearized net applied to pixels with sel==o
__global__ void k_ymask(const float* __restrict__ x, const float* __restrict__ w1,
                        const float* __restrict__ w2, const float* __restrict__ y1,
                        const float* __restrict__ yf, const int* __restrict__ sel,
                        float* __restrict__ ymf) {
  __shared__ float s[CMID][16];
  int blk = blockIdx.x, b = blk >> 6, o = blk & 63;
  int oh = o >> 3, ow = o & 7;
  int tid = threadIdx.x;           // 128 threads
  for (int task = tid; task < CMID * 16; task += 128) {
    int m = task >> 4, slot = task & 15;
    int k2h = slot >> 2, k2w = slot & 3;
    int ph = oh * 2 - 1 + k2h, pw = ow * 2 - 1 + k2w;
    float v = 0.f;
    if ((unsigned)ph < 16u && (unsigned)pw < 16u &&
        y1[((b * CMID + m) * 16 + ph) * 16 + pw] > 0.f) {
      for (int k1 = 0; k1 < 16; ++k1) {
        int k1h = k1 >> 2, k1w = k1 & 3;
        int ih = ph * 2 - 1 + k1h, iw = pw * 2 - 1 + k1w;
        if ((unsigned)ih < 32u && (unsigned)iw < 32u) {
          int i = ih * 32 + iw;
          if (sel[b * 1024 + i] == o) {
            for (int ci = 0; ci < CIN; ++ci)
              v += w1[((m * CIN + ci) * 4 + k1h) * 4 + k1w] *
                   x[((b * CIN + ci) * 32 + ih) * 32 + iw];
          }
        }
      }
    }
    s[m][slot] = v;
  }
  __syncthreads();
  if (tid < COUT) {
    int c = tid;
    float acc = 0.f;
    if (yf[(b * NPOS + o) * COUT + c] > 0.f) {    // relu'(z2)
      for (int m = 0; m < CMID; ++m)
        for (int slot = 0; slot < 16; ++slot)
          acc += w2[((c * CMID + m) * 4 + (slot >> 2)) * 4 + (slot & 3)] * s[m][slot];
    }
    ymf[(b * NPOS + o) * COUT + c] = acc;
  }
}

extern "C" void kernel_launch(void* const* d_in, const int* in_sizes, int n_in,
                              void* d_out, int out_size, void* d_ws, size_t ws_size,
                              hipStream_t stream) {
  const float* x  = (const float*)d_in[0];
  const float* w1 = (const float*)d_in[1];
  const float* b1 = (const float*)d_in[2];
  const float* w2 = (const float*)d_in[3];
  const float* b2 = (const float*)d_in[4];
  const float* Km = (const float*)d_in[5];
  const float* Vm = (const float*)d_in[6];
  float* out = (float*)d_out;

  float* ws  = (float*)d_ws;
  float* y1  = ws;                 // 16384
  float* yf  = y1 + 16384;         // 8192
  float* S   = yf + 8192;          // 65536 (scores/attn, reused twice)
  float* yq  = S + 65536;          // 8192
  float* g2  = yq + 8192;          // 8192
  float* et  = g2 + 8192;          // 131072
  float* ymf = et + 131072;        // 8192
  int*   sel = (int*)(ymf + 8192); // 2048 ints

  const float beta = 0.125f;       // 1/sqrt(64)

  k_conv1  <<<64, 256, 0, stream>>>(x, w1, b1, y1);
  k_conv2  <<<32, 256, 0, stream>>>(y1, w2, b2, yf);
  // q(y)
  k_scores <<<32, 256, 0, stream>>>(yf, Km, S, beta);
  k_softmax<<<16, 256, 0, stream>>>(S, NB * NPOS);
  k_attnv  <<< 4, 256, 0, stream>>>(S, Vm, yq, 0);
  // e_total -> sel
  k_grad   <<<32, 256, 0, stream>>>(yf, yq, g2);
  k_etotal <<<128, 128, 0, stream>>>(g2, w2, y1, w1, et);
  k_argmin <<< 8, 256, 0, stream>>>(et, sel);
  // y_masked
  k_ymask  <<<128, 128, 0, stream>>>(x, w1, w2, y1, yf, sel, ymf);
  // q(y_masked) -> out (NCHW)
  k_scores <<<32, 256, 0, stream>>>(ymf, Km, S, beta);
  k_softmax<<<16, 256, 0, stream>>>(S, NB * NPOS);
  k_attnv  <<< 4, 256, 0, stream>>>(S, Vm, out, 1);
}